// NDConvHierarchical_17033840296158
// MI455X (gfx1250) — compile-verified
//
#include <hip/hip_runtime.h>
#include <hip/hip_bf16.h>

// CDNA5 / gfx1250: wave32, WMMA (not MFMA).
typedef float v2f __attribute__((ext_vector_type(2)));
typedef float v8f __attribute__((ext_vector_type(8)));

#define LDS_PAD 4   // de-phase LDS banks between spatial positions

// Fused: 4-D conv (3^4 taps, pad 1) + bias + ReLU + 2^4 mean-pool.
// One block = one (batch, pooled-position pair along p3). One wave = one
// 16-out-channel tile x TILES pooled positions (weight fragments shared).
// WMMA tile: M = 16 conv positions of one pool block, N = 16 out channels,
// K = Ci contracted in steps of 4 via V_WMMA_F32_16X16X4_F32, 81 taps.
template<int Ci, int Co, int S, int TILES>
__global__ __launch_bounds__((Co/16)*32)
void conv4d_relu_pool_wmma(const float* __restrict__ x,
                           const float* __restrict__ w,
                           const float* __restrict__ bias,
                           float* __restrict__ out)
{
    constexpr int P    = S / 2;
    constexpr int PH   = P / TILES;          // pooled p3 handled in pairs
    constexpr int P4   = P * P * P * P;
    constexpr int D3   = 2 + 2 * TILES;      // patch extent along dim3 (halo incl.)
    constexpr int NSP  = 64 * D3;            // patch spatial positions
    constexpr int CSTR = Ci + LDS_PAD;
    constexpr int NT   = (Co / 16) * 32;

    __shared__ float lds[NSP * CSTR];        // [d0][d1][d2][d3] x Ci, channel-minor

    const int b   = blockIdx.y;
    const int bx  = blockIdx.x;              // ((p0*P+p1)*P+p2)*PH + p3h
    const int p3h = bx % PH;
    const int p2  = (bx / PH) % P;
    const int p1  = (bx / (PH * P)) % P;
    const int p0  = bx / (PH * P * P);

    // ---- stage 4*4*4*D3 x Ci padded input patch into LDS ----
    const int tid = threadIdx.x;
    for (int e = tid; e < NSP * Ci; e += NT) {
        const int ci = e & (Ci - 1);
        const int sp = e / Ci;
        const int d3 = sp % D3;
        const int r  = sp / D3;
        const int d2 = r & 3, d1 = (r >> 2) & 3, d0 = (r >> 4) & 3;
        const int x0 = 2 * p0 - 1 + d0;
        const int x1 = 2 * p1 - 1 + d1;
        const int x2 = 2 * p2 - 1 + d2;
        const int x3 = 2 * (p3h * TILES) - 1 + d3;
        float v = 0.0f;
        if ((unsigned)x0 < (unsigned)S && (unsigned)x1 < (unsigned)S &&
            (unsigned)x2 < (unsigned)S && (unsigned)x3 < (unsigned)S) {
            v = x[((((b * Ci + ci) * S + x0) * S + x1) * S + x2) * S + x3];
        }
        lds[sp * CSTR + ci] = v;
    }
    __syncthreads();

    // ---- implicit GEMM on the matrix pipe ----
    const int lane = tid & 31;
    const int wave = tid >> 5;
    const int n    = lane & 15;              // N (out-channel) index; also M index
    const int klo  = (lane < 16) ? 0 : 2;    // K split across half-waves (ISA A/B layout)
    // M lane -> pool-block offset bits (order irrelevant: pooled mean over M)
    const int qsp  = (((n >> 3) & 1) * 16 + ((n >> 2) & 1) * 4 + ((n >> 1) & 1)) * D3 + (n & 1);
    const int coA  = wave * 16 + n;          // absolute output channel
    const int wCoK = coA * Ci * 81 + klo * 81;

    v8f acc[TILES];
    #pragma unroll
    for (int u = 0; u < TILES; ++u)
        acc[u] = (v8f){0.f, 0.f, 0.f, 0.f, 0.f, 0.f, 0.f, 0.f};

    #pragma unroll 1
    for (int t0 = 0; t0 < 3; ++t0) {
        #pragma unroll 1
        for (int t1 = 0; t1 < 3; ++t1) {
            #pragma unroll 1
            for (int t2 = 0; t2 < 3; ++t2) {
                // A base row in LDS and B base row in W for this tap triple
                const int row  = (qsp + (t0 * 16 + t1 * 4 + t2) * D3) * CSTR + klo;
                const int wRow = wCoK + t0 * 27 + t1 * 9 + t2 * 3;
                #pragma unroll 4
                for (int kb = 0; kb < Ci; kb += 4) {
                    // 3 consecutive taps of weights: merge into b96 loads
                    const float* wp = &w[wRow + kb * 81];
                    float wx[3] = { wp[0],  wp[1],  wp[2]  };   // ci = kb+klo
                    float wy[3] = { wp[81], wp[82], wp[83] };   // ci = kb+klo+1
                    #pragma unroll
                    for (int t3 = 0; t3 < 3; ++t3) {
                        v2f bf; bf.x = wx[t3]; bf.y = wy[t3];
                        #pragma unroll
                        for (int u = 0; u < TILES; ++u) {
                            v2f a = *(const v2f*)&lds[row + (t3 + 2 * u) * CSTR + kb];
                            acc[u] = __builtin_amdgcn_wmma_f32_16x16x4_f32(
                                /*neg_a=*/false, a, /*neg_b=*/false, bf,
                                /*c_mod=*/(short)0, acc[u],
                                /*reuse_a=*/false, /*reuse_b=*/false);
                        }
                    }
                }
            }
        }
    }

    // ---- epilogue: bias + ReLU per conv output, mean over the 16 M rows ----
    const float bs = bias[coA];
    #pragma unroll
    for (int u = 0; u < TILES; ++u) {
        float s = 0.0f;
        #pragma unroll
        for (int r = 0; r < 8; ++r) {
            float v = acc[u][r] + bs;        // C VGPR r: M=r (lo half) / M=r+8 (hi half)
            s += (v > 0.0f) ? v : 0.0f;
        }
        s += __shfl_xor(s, 16, 32);          // combine M and M+8 halves (wave32)
        if (lane < 16) {
            const int pIdx = ((p0 * P + p1) * P + p2) * P + p3h * TILES + u;
            out[(b * Co + coA) * P4 + pIdx] = s * (1.0f / 16.0f);
        }
    }
}

extern "C" void kernel_launch(void* const* d_in, const int* in_sizes, int n_in,
                              void* d_out, int out_size, void* d_ws, size_t ws_size,
                              hipStream_t stream) {
    (void)in_sizes; (void)n_in; (void)out_size; (void)ws_size;

    const float* src = (const float*)d_in[0];
    const float* W0  = (const float*)d_in[1];
    const float* b0  = (const float*)d_in[2];
    const float* W1  = (const float*)d_in[3];
    const float* b1  = (const float*)d_in[4];
    const float* W2  = (const float*)d_in[5];
    const float* b2  = (const float*)d_in[6];
    const float* W3  = (const float*)d_in[7];
    const float* b3  = (const float*)d_in[8];

    float* t0 = (float*)d_ws;                  // [8,32,8,8,8,8]  = 4 MB scratch
    float* o1 = (float*)d_out;                 // saved[0]: [8,64,4^4] = 131072
    float* o2 = o1 + 8 * 64 * 256;             // saved[1]: [8,64,2^4] = 8192
    float* o3 = o2 + 8 * 64 * 16;              // saved[2]: [8,64,1]   = 512

    // L0: Ci=8,  Co=32, S=16 -> pooled 8^4 ; 2 waves/block, 2 tiles/wave
    conv4d_relu_pool_wmma<8, 32, 16, 2><<<dim3(2048, 8), 64, 0, stream>>>(src, W0, b0, t0);
    // L1: Ci=32, Co=64, S=8  -> pooled 4^4 ; 4 waves/block, 2 tiles/wave
    conv4d_relu_pool_wmma<32, 64, 8, 2><<<dim3(128, 8), 128, 0, stream>>>(t0, W1, b1, o1);
    // L2: Ci=64, Co=64, S=4  -> pooled 2^4 ; 2 tiles/wave
    conv4d_relu_pool_wmma<64, 64, 4, 2><<<dim3(8, 8), 128, 0, stream>>>(o1, W2, b2, o2);
    // L3: Ci=64, Co=64, S=2  -> pooled 1  ; single tile
    conv4d_relu_pool_wmma<64, 64, 2, 1><<<dim3(1, 8), 128, 0, stream>>>(o2, W3, b3, o3);
}